// adder2d2_0_63926293233871
// MI455X (gfx1250) — compile-verified
//
#include <hip/hip_runtime.h>
#include <stdint.h>

// AdderNet 3x3 conv, stride 1, pad 1.
// x: [8,64,32,32] f32, W: [64,64,3,3] f32, out: [8,64,32,32] f32
// out[n,f,y,x] = -sum_{c,kh,kw} |W[f,c,kh,kw] - xpad[n,c,y+kh-1,x+kw-1]|
//
// Compute-bound VALU kernel (~604M sub/abs-add ops; memory floor ~4 MB =>
// ~180 ns at 23.3 TB/s -- everything lives in the 192 MB L2). WMMA is
// inapplicable: L1 distance is not bilinear and the thermometer-code matmul
// rewrite inflates work/data 255x. CDNA5-specific paths used:
//  - async global->LDS DMA (global_load_async_to_lds_b128 / s_wait_asynccnt),
//    double-buffered so channel c+1's copy overlaps channel c's compute
//  - weights fetched as SMEM s_loads (constant addrspace) into SGPRs,
//    issued under the DMA wait.

#define TILE_STRIDE 40           // floats per row (interior starts at col 4 -> 16B aligned)
#define TILE_ROWS   10
#define TILE_ELEMS  (TILE_ROWS * TILE_STRIDE)

// Read-only (constant addrspace) load => backend selects scalar s_load when
// the address is uniform. Cast via integer so clang accepts it everywhere.
__device__ __forceinline__ float scalar_ld(const float* p) {
    return *(const __attribute__((address_space(4))) float*)(uintptr_t)p;
}

__global__ __launch_bounds__(256) void adder2d_kernel(
    const float* __restrict__ x,
    const float* __restrict__ W,
    float* __restrict__ out)
{
    __shared__ __align__(16) float tile[2][TILE_ELEMS];

    const int tid = threadIdx.x;
    const int id  = blockIdx.x;
    const int yb  = id & 3;          // row band (8 rows each)
    const int fq  = (id >> 2) & 15;  // filter quad
    const int n   = id >> 6;         // batch index
    const int col = tid & 31;        // output x
    const int r   = tid >> 5;        // row within band (0..7)
    const int y   = yb * 8 + r;      // output y
    const int f0  = fq * 4;

    // ---- zero both tile buffers with one b128 store (halo stays zero) ----
    if (tid < (2 * TILE_ELEMS) / 4) {
        ((float4*)&tile[0][0])[tid] = make_float4(0.f, 0.f, 0.f, 0.f);
    }

    // ---- async-copy lane setup: threads 0..79 -> (row tr, 16B segment seg) ----
    const int  tr     = tid >> 3;              // tile row (only <10 used)
    const int  seg    = tid & 7;               // 0..7  (8 x b128 = 32 floats)
    const int  gy     = yb * 8 + tr - 1;       // input row for tile row tr
    const bool issuer = (tid < 80) && (gy >= 0) && (gy < 32);

    const unsigned lds0 = (unsigned)(uintptr_t)(&tile[0][tr * TILE_STRIDE + 4 + seg * 4]);
    const unsigned lds1 = (unsigned)(uintptr_t)(&tile[1][tr * TILE_STRIDE + 4 + seg * 4]);
    const unsigned goff = (unsigned)((gy * 32 + seg * 4) * 4);   // byte offset in channel image
    const float* xn = x + (size_t)n * 64 * 1024;                 // + c*1024 per channel

    __syncthreads();   // zeros visible before any DMA lands

    // ---- preload channel 0 into buffer 0 ----
    if (issuer) {
        unsigned long long base = (unsigned long long)(uintptr_t)xn;
        asm volatile("global_load_async_to_lds_b128 %0, %1, %2"
                     :: "v"(lds0), "v"(goff), "s"(base) : "memory");
    }

    // negative accumulation: out = acc directly, no final negate
    float acc0 = 0.f, acc1 = 0.f, acc2 = 0.f, acc3 = 0.f;
    float w0[9], w1[9], w2[9], w3[9];

    // uniform weight base; per-filter stride = 64*9 = 576 elements
    const int wbase0 = __builtin_amdgcn_readfirstlane((f0 * 64) * 9);

    #define LOAD_WEIGHTS(cc)                                                   \
        do {                                                                   \
            const float* wp = W + wbase0 + (cc) * 9;                           \
            _Pragma("unroll")                                                  \
            for (int k = 0; k < 9; ++k) {                                      \
                w0[k] = scalar_ld(wp + k);                                     \
                w1[k] = scalar_ld(wp + 576 + k);                               \
                w2[k] = scalar_ld(wp + 1152 + k);                              \
                w3[k] = scalar_ld(wp + 1728 + k);                              \
            }                                                                  \
        } while (0)

    #define COMPUTE(tb)                                                        \
        do {                                                                   \
            _Pragma("unroll")                                                  \
            for (int kh = 0; kh < 3; ++kh) {                                   \
                _Pragma("unroll")                                              \
                for (int kw = 0; kw < 3; ++kw) {                               \
                    const int   k  = kh * 3 + kw;                              \
                    const float xv = (tb)[(r + kh) * TILE_STRIDE + (col + kw + 3)]; \
                    acc0 -= fabsf(w0[k] - xv);                                 \
                    acc1 -= fabsf(w1[k] - xv);                                 \
                    acc2 -= fabsf(w2[k] - xv);                                 \
                    acc3 -= fabsf(w3[k] - xv);                                 \
                }                                                              \
            }                                                                  \
        } while (0)

    // ---- main loop: issue DMA(c+1), fetch weights(c) under the DMA wait ----
    for (int c = 0; c < 63; ++c) {
        if (issuer) {
            unsigned long long base =
                (unsigned long long)(uintptr_t)(xn + (size_t)(c + 1) * 1024);
            unsigned ldsb = ((c + 1) & 1) ? lds1 : lds0;
            asm volatile("global_load_async_to_lds_b128 %0, %1, %2"
                         :: "v"(ldsb), "v"(goff), "s"(base) : "memory");
        }
        LOAD_WEIGHTS(c);                  // s_loads fly while we wait on DMA
        asm volatile("s_wait_asynccnt 1" ::: "memory");
        __syncthreads();                  // channel-c tile visible to all waves

        COMPUTE(tile[c & 1]);

        __syncthreads();                  // all reads of buf[c&1] done before refill
    }

    // ---- peeled last channel (c = 63, buffer 1) ----
    LOAD_WEIGHTS(63);
    asm volatile("s_wait_asynccnt 0" ::: "memory");
    __syncthreads();
    COMPUTE(tile[1]);

    const size_t ob = (((size_t)n * 64 + f0) * 32 + y) * 32 + col;
    out[ob]          = acc0;
    out[ob + 1024]   = acc1;
    out[ob + 2048]   = acc2;
    out[ob + 3072]   = acc3;

    #undef LOAD_WEIGHTS
    #undef COMPUTE
}

extern "C" void kernel_launch(void* const* d_in, const int* in_sizes, int n_in,
                              void* d_out, int out_size, void* d_ws, size_t ws_size,
                              hipStream_t stream) {
    const float* x = (const float*)d_in[0];
    const float* W = (const float*)d_in[1];
    float* out     = (float*)d_out;
    // 8 batches * 16 filter-quads * 4 row-bands = 512 workgroups
    adder2d_kernel<<<dim3(512), dim3(256), 0, stream>>>(x, W, out);
}